// MultiHeadAttentionLayer_54700703482419
// MI455X (gfx1250) — compile-verified
//
#include <hip/hip_runtime.h>
#include <hip/hip_bf16.h>

typedef __attribute__((ext_vector_type(2))) float v2f;
typedef __attribute__((ext_vector_type(8))) float v8f;

#define N_V   100000
#define N_E   20000
#define N_NZ  400000
#define CIN   256
#define HID   256
#define NH    4
#define DH    64
#define NEG_SLOPE 0.2f

// ---------------------------------------------------------------------------
// Stage 1: v2e scatter-sum of raw X into per-edge sums + membership counts.
// One wave per incidence; float4 (b128) row loads, scalar f32 atomics out.
// ---------------------------------------------------------------------------
__global__ __launch_bounds__(256) void v2e_scatter(
    const float* __restrict__ X, const int* __restrict__ inc_v,
    const int* __restrict__ inc_e, float* __restrict__ Xsum,
    float* __restrict__ cnt)
{
  const int widx = blockIdx.x * 8 + (threadIdx.x >> 5);
  if (widx >= N_NZ) return;
  const int lane = threadIdx.x & 31;
  const int v = inc_v[widx];
  const int e = inc_e[widx];
  const float4* __restrict__ src4 = (const float4*)(X + (size_t)v * CIN);
  float* __restrict__ dst = Xsum + (size_t)e * HID;
#pragma unroll
  for (int f4 = lane; f4 < HID / 4; f4 += 32) {
    const float4 val = src4[f4];
    atomicAdd(&dst[f4 * 4 + 0], val.x);
    atomicAdd(&dst[f4 * 4 + 1], val.y);
    atomicAdd(&dst[f4 * 4 + 2], val.z);
    atomicAdd(&dst[f4 * 4 + 3], val.w);
  }
  if (lane == 0) atomicAdd(&cnt[e], 1.0f);
}

// ---------------------------------------------------------------------------
// Stage 2: Y[e] = (Xsum[e] / max(cnt,1)) @ Wcat + b   via V_WMMA_F32_16X16X4_F32
// wave -> (16-row edge tile, one head = 64 output cols = 4 WMMA col tiles).
// Mean division fused as a per-row scale of the A fragment (commutes with
// the matmul; bias added at store).
// ---------------------------------------------------------------------------
__global__ __launch_bounds__(256) void gemm_edge(
    const float* __restrict__ Xsum, const float* __restrict__ cnt,
    const float* __restrict__ W, const float* __restrict__ bvec,
    float* __restrict__ Y)
{
  const int lane = threadIdx.x & 31;
  const int wid  = blockIdx.x * 8 + (threadIdx.x >> 5);   // 0..4999
  const int row_tile = wid >> 2;                          // 0..1249
  const int head = wid & 3;                               // 0..3
  const int row0 = row_tile * 16;
  const int m = lane & 15;
  const int half = lane >> 4;

  const int row = row0 + m;
  const float inv = 1.0f / fmaxf(cnt[row], 1.0f);
  const float* __restrict__ arow = Xsum + (size_t)row * HID;
  const float* __restrict__ wh = W + (size_t)head * (CIN * DH);

  v8f acc[4] = {v8f{}, v8f{}, v8f{}, v8f{}};
#pragma unroll 4
  for (int k0 = 0; k0 < CIN; k0 += 4) {
    const int ka = k0 + 2 * half;            // this half-wave's K pair
    v2f afrag;
    afrag.x = arow[ka] * inv;
    afrag.y = arow[ka + 1] * inv;
    const float* __restrict__ wk = wh + (size_t)ka * DH + m;
#pragma unroll
    for (int t = 0; t < 4; ++t) {
      v2f bfrag;
      bfrag.x = wk[t * 16];                  // W[head][ka  ][t*16+m]
      bfrag.y = wk[t * 16 + DH];             // W[head][ka+1][t*16+m]
      acc[t] = __builtin_amdgcn_wmma_f32_16x16x4_f32(
          false, afrag, false, bfrag, (short)0, acc[t], false, false);
    }
  }

  // C/D layout: VGPR r -> (M = half*8 + r, N = lane&15)
  const int mbase = half * 8;
#pragma unroll
  for (int t = 0; t < 4; ++t) {
    const int col = head * DH + t * 16 + m;
    const float bias = bvec[col];
#pragma unroll
    for (int r = 0; r < 8; ++r) {
      Y[(size_t)(row0 + mbase + r) * HID + col] = acc[t][r] + bias;
    }
  }
}

// ---------------------------------------------------------------------------
// Stage 3: es[e,h] = exp(clip(leaky(Y[e,h,:]·a_e[h,:]), 0.001, 5)).
// Since scores are clamped to [0.001, 5], exp(s) in [1.001, 148.4]: the
// softmax max-subtraction is unnecessary, so we store exp(s) directly and
// skip the whole segment-max pass.
// ---------------------------------------------------------------------------
__global__ __launch_bounds__(256) void edge_alpha(
    const float* __restrict__ Y, const float* __restrict__ a_e,
    float* __restrict__ es)
{
  __shared__ float red[256];
  const int e = blockIdx.x;
  const int f = threadIdx.x;
  red[f] = Y[(size_t)e * HID + f] * a_e[f];
  __syncthreads();
#pragma unroll
  for (int off = 32; off >= 1; off >>= 1) {
    if ((f & 63) < off) red[f] += red[f + off];
    __syncthreads();
  }
  if ((f & 63) == 0) {
    const float alpha = red[f];
    float s = alpha >= 0.0f ? alpha : NEG_SLOPE * alpha;
    s = fminf(fmaxf(s, 0.001f), 5.0f);
    es[e * NH + (f >> 6)] = __expf(s);
  }
}

// ---------------------------------------------------------------------------
// Stage 4: softmax denominators: denom[v,h] += es[e,h].
// One thread per incidence: single b128 gather of 4 per-head exp-scores.
// ---------------------------------------------------------------------------
__global__ __launch_bounds__(256) void vert_denom(
    const float* __restrict__ es, const int* __restrict__ inc_v,
    const int* __restrict__ inc_e, float* __restrict__ denom)
{
  const int inc = blockIdx.x * 256 + threadIdx.x;
  if (inc >= N_NZ) return;
  const int v = inc_v[inc], e = inc_e[inc];
  const float4 s4 = ((const float4*)es)[e];
  float* __restrict__ d = denom + (size_t)v * NH;
  atomicAdd(&d[0], s4.x);
  atomicAdd(&d[1], s4.y);
  atomicAdd(&d[2], s4.z);
  atomicAdd(&d[3], s4.w);
}

// ---------------------------------------------------------------------------
// Stage 5: out[v] += (es[e,h]/denom[v,h]) * Y[e].  Wave per incidence,
// b128 row loads, scalar f32 atomics to the destination row.
// ---------------------------------------------------------------------------
__global__ __launch_bounds__(256) void aggregate(
    const float* __restrict__ Y, const float* __restrict__ es,
    const int* __restrict__ inc_v, const int* __restrict__ inc_e,
    const float* __restrict__ denom, float* __restrict__ out)
{
  const int widx = blockIdx.x * 8 + (threadIdx.x >> 5);
  if (widx >= N_NZ) return;
  const int lane = threadIdx.x & 31;
  const int v = inc_v[widx];
  const int e = inc_e[widx];
  const float4 s4 = ((const float4*)es)[e];
  const float4 d4 = ((const float4*)denom)[v];
  float wgt[NH] = { s4.x / d4.x, s4.y / d4.y, s4.z / d4.z, s4.w / d4.w };

  const float4* __restrict__ yrow4 = (const float4*)(Y + (size_t)e * HID);
  float* __restrict__ orow = out + (size_t)v * HID;
#pragma unroll
  for (int f4 = lane; f4 < HID / 4; f4 += 32) {
    const float w = wgt[f4 >> 4];            // 16 float4s per head
    const float4 y = yrow4[f4];
    atomicAdd(&orow[f4 * 4 + 0], w * y.x);
    atomicAdd(&orow[f4 * 4 + 1], w * y.y);
    atomicAdd(&orow[f4 * 4 + 2], w * y.z);
    atomicAdd(&orow[f4 * 4 + 3], w * y.w);
  }
}

// Stage 6: ELU, float4-vectorized.
__global__ __launch_bounds__(256) void elu_kernel(float4* __restrict__ out)
{
  const size_t i = (size_t)blockIdx.x * 256 + threadIdx.x;
  if (i >= (size_t)N_V * HID / 4) return;
  float4 x = out[i];
  x.x = x.x > 0.0f ? x.x : expm1f(x.x);
  x.y = x.y > 0.0f ? x.y : expm1f(x.y);
  x.z = x.z > 0.0f ? x.z : expm1f(x.z);
  x.w = x.w > 0.0f ? x.w : expm1f(x.w);
  out[i] = x;
}

// ---------------------------------------------------------------------------
extern "C" void kernel_launch(void* const* d_in, const int* in_sizes, int n_in,
                              void* d_out, int out_size, void* d_ws, size_t ws_size,
                              hipStream_t stream) {
  const float* X     = (const float*)d_in[0];   // [N, 256]
  const float* W     = (const float*)d_in[1];   // [H, 256, 64]
  const float* bvec  = (const float*)d_in[2];   // [H, 64]
  const float* a_e   = (const float*)d_in[3];   // [H, 64]
  const int*   inc_v = (const int*)d_in[4];     // [NNZ]
  const int*   inc_e = (const int*)d_in[5];     // [NNZ]
  float* out = (float*)d_out;                   // [N, 256]

  float* ws    = (float*)d_ws;
  float* Xsum  = ws;                                  // E*256
  float* Y     = Xsum + (size_t)N_E * HID;            // E*256
  float* cnt   = Y    + (size_t)N_E * HID;            // E
  float* es    = cnt  + N_E;                          // E*4  (16B aligned)
  float* denom = es   + (size_t)N_E * NH;             // N*4  (16B aligned)

  hipMemsetAsync(Xsum, 0, sizeof(float) * (size_t)N_E * HID, stream);
  hipMemsetAsync(cnt, 0, sizeof(float) * N_E, stream);
  hipMemsetAsync(denom, 0, sizeof(float) * (size_t)N_V * NH, stream);
  hipMemsetAsync(out, 0, sizeof(float) * (size_t)N_V * HID, stream);

  v2e_scatter<<<dim3(N_NZ / 8), 256, 0, stream>>>(X, inc_v, inc_e, Xsum, cnt);
  gemm_edge<<<dim3((N_E / 16) * NH / 8), 256, 0, stream>>>(Xsum, cnt, W, bvec, Y);
  edge_alpha<<<dim3(N_E), 256, 0, stream>>>(Y, a_e, es);
  vert_denom<<<dim3((N_NZ + 255) / 256), 256, 0, stream>>>(es, inc_v, inc_e, denom);
  aggregate<<<dim3(N_NZ / 8), 256, 0, stream>>>(Y, es, inc_v, inc_e, denom, out);
  elu_kernel<<<dim3(((size_t)N_V * HID / 4 + 255) / 256), 256, 0, stream>>>((float4*)out);
}